// SageGNN_model_5927054868537
// MI455X (gfx1250) — compile-verified
//
#include <hip/hip_runtime.h>
#include <hip/hip_bf16.h>

typedef __attribute__((ext_vector_type(2))) float v2f;
typedef __attribute__((ext_vector_type(8))) float v8f;

namespace {
constexpr int NN = 50000;
constexpr int EE = 800000;
constexpr int BB = 128;
constexpr int DD = 64;
constexpr int HH = 128;
constexpr int CC = 2;
constexpr float EPSV = 1e-5f;
}

__device__ __forceinline__ v8f zero8() {
    v8f c;
#pragma unroll
    for (int i = 0; i < 8; ++i) c[i] = 0.f;
    return c;
}

// One wave computes a 16x16 fp32 tile of A(16xK) @ B(Kx16) via chained
// V_WMMA_F32_16X16X4_F32. At = A tile base (row-major, lda), Bt = B tile base
// (row-major, ldb), K multiple of 4.
// A frag layout: lanes 0-15 hold rows M=lane, v0=K0 v1=K1; lanes 16-31 K2/K3.
// B frag mirrors with columns.
__device__ __forceinline__ v8f wmma16x16_f32(const float* __restrict__ At, int lda,
                                             const float* __restrict__ Bt, int ldb,
                                             v8f c, int lane) {
    const int half = lane >> 4;
    const int idx  = lane & 15;
#pragma unroll
    for (int k = 0; k < DD; k += 4) {
        const int k0 = k + half * 2;
        v2f a, b;
        a.x = At[idx * lda + k0];
        a.y = At[idx * lda + k0 + 1];
        b.x = Bt[k0 * ldb + idx];
        b.y = Bt[(k0 + 1) * ldb + idx];
        c = __builtin_amdgcn_wmma_f32_16x16x4_f32(false, a, false, b, (short)0, c,
                                                  false, false);
    }
    return c;
}

// ---------------- utility fills ----------------
__global__ void k_fill_f32(float* p, int n, float v) {
    int i = blockIdx.x * blockDim.x + threadIdx.x;
    if (i < n) p[i] = v;
}
__global__ void k_fill_u32(unsigned* p, int n, unsigned v) {
    int i = blockIdx.x * blockDim.x + threadIdx.x;
    if (i < n) p[i] = v;
}

// ---------------- edge scatter: agg[dst] += x[src], cnt[dst] += 1 ----------------
__global__ void k_scatter(const float* __restrict__ x, const int* __restrict__ ei,
                          float* __restrict__ agg, float* __restrict__ cnt) {
    unsigned i = blockIdx.x * blockDim.x + threadIdx.x;
    if (i >= (unsigned)EE * DD) return;
    unsigned e = i >> 6;
    unsigned d = i & 63u;
    int src = ei[e];
    int dst = ei[EE + e];
    atomicAdd(&agg[(size_t)dst * DD + d], x[(size_t)src * DD + d]);
    if (d == 0) atomicAdd(&cnt[dst], 1.0f);
}

// ---------------- mean = agg / max(cnt,1) (in place) ----------------
__global__ void k_mean(float* __restrict__ agg, const float* __restrict__ cnt) {
    int i = blockIdx.x * blockDim.x + threadIdx.x;
    if (i >= NN * DD) return;
    int n = i >> 6;
    agg[i] = agg[i] / fmaxf(cnt[n], 1.0f);
}

// ---------------- hpre = mean@W_l + x@W_r + b_l ; xres = x@W_res ----------------
__global__ void k_linear(const float* __restrict__ mean, const float* __restrict__ x,
                         const float* __restrict__ Wl, const float* __restrict__ bl,
                         const float* __restrict__ Wr, const float* __restrict__ Wres,
                         float* __restrict__ hpre, float* __restrict__ xres) {
    const int wave = blockIdx.x * (blockDim.x >> 5) + (threadIdx.x >> 5);
    const int lane = threadIdx.x & 31;
    const int total = (NN / 16) * (DD / 16);
    if (wave >= total) return;
    const int tm = wave >> 2;          // node tile
    const int tn = wave & 3;           // column tile
    const int m0 = tm * 16, n0 = tn * 16;

    v8f c1 = zero8();
    c1 = wmma16x16_f32(mean + (size_t)m0 * DD, DD, Wl + n0, DD, c1, lane);
    c1 = wmma16x16_f32(x + (size_t)m0 * DD, DD, Wr + n0, DD, c1, lane);
    v8f c2 = zero8();
    c2 = wmma16x16_f32(x + (size_t)m0 * DD, DD, Wres + n0, DD, c2, lane);

    const int half = lane >> 4, idx = lane & 15;
    const float bv = bl[n0 + idx];
#pragma unroll
    for (int i = 0; i < 8; ++i) {
        const int r = m0 + i + half * 8;
        hpre[(size_t)r * DD + n0 + idx] = c1[i] + bv;
        xres[(size_t)r * DD + n0 + idx] = c2[i];
    }
}

// ---------------- column sum / sumsq of hpre ----------------
__global__ void k_colstats(const float* __restrict__ hpre, float* __restrict__ stats) {
    const int t = threadIdx.x;          // 256
    const int col = t & 63;
    const int r0 = blockIdx.x * 4 + (t >> 6);
    float s = 0.f, ss = 0.f;
    for (int r = r0; r < NN; r += gridDim.x * 4) {
        float v = hpre[(size_t)r * DD + col];
        s += v;
        ss += v * v;
    }
    atomicAdd(&stats[col], s);
    atomicAdd(&stats[DD + col], ss);
}

// ---------------- batchnorm scale/shift ----------------
__global__ void k_normfin(const float* __restrict__ stats, const float* __restrict__ gamma,
                          const float* __restrict__ beta, float* __restrict__ normA,
                          float* __restrict__ normB) {
    int d = threadIdx.x;
    if (d >= DD) return;
    float mu = stats[d] / (float)NN;
    float var = stats[DD + d] / (float)NN - mu * mu;
    float inv = rsqrtf(var + EPSV);
    float a = gamma[d] * inv;
    normA[d] = a;
    normB[d] = beta[d] - mu * a;
}

// ---------------- h = hpre*normA + normB + xres + b_res ----------------
__global__ void k_hfinal(const float* __restrict__ hpre, const float* __restrict__ xres,
                         const float* __restrict__ normA, const float* __restrict__ normB,
                         const float* __restrict__ bres, float* __restrict__ h) {
    int i = blockIdx.x * blockDim.x + threadIdx.x;
    if (i >= NN * DD) return;
    int d = i & 63;
    h[i] = hpre[i] * normA[d] + normB[d] + xres[i] + bres[d];
}

// ---------------- g = relu(h@gW1+gb1)@gW2 + gb2, one wave per 16 nodes ----------------
__global__ void k_gate(const float* __restrict__ h, const float* __restrict__ gW1,
                       const float* __restrict__ gb1, const float* __restrict__ gW2,
                       const float* __restrict__ gb2, float* __restrict__ g) {
    const int wave = blockIdx.x * (blockDim.x >> 5) + (threadIdx.x >> 5);
    const int lane = threadIdx.x & 31;
    if (wave >= NN / 16) return;
    const int m0 = wave * 16;
    const int half = lane >> 4, idx = lane & 15;

    float dp[8];
#pragma unroll
    for (int i = 0; i < 8; ++i) dp[i] = 0.f;

    for (int j = 0; j < DD / 16; ++j) {
        const int n0 = j * 16;
        v8f c = zero8();
        c = wmma16x16_f32(h + (size_t)m0 * DD, DD, gW1 + n0, DD, c, lane);
        const float b1 = gb1[n0 + idx];
        const float w2 = gW2[n0 + idx];   // gate_W2 is (D,1)
#pragma unroll
        for (int i = 0; i < 8; ++i) {
            float v = fmaxf(c[i] + b1, 0.f);
            dp[i] += v * w2;
        }
    }
    // half-wave (16-lane) reduction: rows 0-7 live in lanes 0-15, rows 8-15 in 16-31
#pragma unroll
    for (int off = 8; off >= 1; off >>= 1) {
#pragma unroll
        for (int i = 0; i < 8; ++i) dp[i] += __shfl_xor(dp[i], off, 32);
    }
    if (idx == 0) {
        const float b2 = gb2[0];
#pragma unroll
        for (int i = 0; i < 8; ++i) g[m0 + half * 8 + i] = dp[i] + b2;
    }
}

// ---------------- order-preserving float<->uint for atomicMax ----------------
__device__ __forceinline__ unsigned enc_f32(float f) {
    unsigned b = __float_as_uint(f);
    return (b & 0x80000000u) ? ~b : (b | 0x80000000u);
}
__device__ __forceinline__ float dec_f32(unsigned u) {
    unsigned b = (u & 0x80000000u) ? (u & 0x7FFFFFFFu) : ~u;
    return __uint_as_float(b);
}

__global__ void k_gmax(const float* __restrict__ g, const int* __restrict__ batch,
                       unsigned* __restrict__ gmax_u) {
    int n = blockIdx.x * blockDim.x + threadIdx.x;
    if (n >= NN) return;
    atomicMax(&gmax_u[batch[n]], enc_f32(g[n]));
}

__global__ void k_gmaxdec(const unsigned* __restrict__ gmax_u, float* __restrict__ gmaxf) {
    int b = threadIdx.x;
    if (b >= BB) return;
    float v = dec_f32(gmax_u[b]);
    gmaxf[b] = isfinite(v) ? v : 0.0f;
}

__global__ void k_zsum(const float* __restrict__ g, const int* __restrict__ batch,
                       const float* __restrict__ gmaxf, float* __restrict__ z) {
    int n = blockIdx.x * blockDim.x + threadIdx.x;
    if (n >= NN) return;
    int b = batch[n];
    atomicAdd(&z[b], expf(g[n] - gmaxf[b]));
}

__global__ void k_pool(const float* __restrict__ g, const int* __restrict__ batch,
                       const float* __restrict__ gmaxf, const float* __restrict__ z,
                       const float* __restrict__ h, float* __restrict__ pooled) {
    int i = blockIdx.x * blockDim.x + threadIdx.x;
    if (i >= NN * DD) return;
    int n = i >> 6;
    int d = i & 63;
    int b = batch[n];
    float alpha = expf(g[n] - gmaxf[b]) / fmaxf(z[b], 1e-16f);
    atomicAdd(&pooled[b * DD + d], alpha * h[i]);
}

// ---------------- t1 = relu(pooled @ W_p1 + b_p1), 128x128, WMMA ----------------
__global__ void k_p1(const float* __restrict__ pooled, const float* __restrict__ Wp1,
                     const float* __restrict__ bp1, float* __restrict__ t1) {
    const int wave = blockIdx.x * (blockDim.x >> 5) + (threadIdx.x >> 5);
    const int lane = threadIdx.x & 31;
    if (wave >= (BB / 16) * (HH / 16)) return;
    const int tm = wave >> 3, tn = wave & 7;
    const int m0 = tm * 16, n0 = tn * 16;
    v8f c = zero8();
    c = wmma16x16_f32(pooled + (size_t)m0 * DD, DD, Wp1 + n0, HH, c, lane);
    const int half = lane >> 4, idx = lane & 15;
    const float bv = bp1[n0 + idx];
#pragma unroll
    for (int i = 0; i < 8; ++i) {
        int r = m0 + i + half * 8;
        t1[(size_t)r * HH + n0 + idx] = fmaxf(c[i] + bv, 0.f);
    }
}

// ---------------- out = t1 @ W_p2 + b_p2, 128x2 ----------------
__global__ void k_p2(const float* __restrict__ t1, const float* __restrict__ Wp2,
                     const float* __restrict__ bp2, float* __restrict__ out) {
    int t = threadIdx.x;       // 256 threads, one block
    if (t >= BB * CC) return;
    int r = t >> 1, c = t & 1;
    float s = 0.f;
    for (int k = 0; k < HH; ++k) s += t1[r * HH + k] * Wp2[k * CC + c];
    out[r * CC + c] = s + bp2[c];
}

extern "C" void kernel_launch(void* const* d_in, const int* in_sizes, int n_in,
                              void* d_out, int out_size, void* d_ws, size_t ws_size,
                              hipStream_t stream) {
    const float* x     = (const float*)d_in[0];
    const int*   ei    = (const int*)d_in[1];
    const int*   batch = (const int*)d_in[2];
    const float* Wl    = (const float*)d_in[3];
    const float* bl    = (const float*)d_in[4];
    const float* Wr    = (const float*)d_in[5];
    const float* Wres  = (const float*)d_in[6];
    const float* bres  = (const float*)d_in[7];
    const float* gamma = (const float*)d_in[8];
    const float* beta  = (const float*)d_in[9];
    const float* gW1   = (const float*)d_in[10];
    const float* gb1   = (const float*)d_in[11];
    const float* gW2   = (const float*)d_in[12];
    const float* gb2   = (const float*)d_in[13];
    const float* Wp1   = (const float*)d_in[14];
    const float* bp1   = (const float*)d_in[15];
    const float* Wp2   = (const float*)d_in[16];
    const float* bp2   = (const float*)d_in[17];
    float* out = (float*)d_out;

    // ---- workspace layout (floats) ----
    const size_t ND = (size_t)NN * DD;
    const size_t NPAD = ((NN + 63) / 64) * 64;
    float* agg   = (float*)d_ws;
    float* cnt   = agg + ND;
    float* hpre  = cnt + NPAD;
    float* xres  = hpre + ND;
    float* h     = xres + ND;
    float* g     = h + ND;
    float* stats = g + NPAD;           // colsum[64] | colsumsq[64]
    float* normA = stats + 2 * DD;
    float* normB = normA + DD;
    float* gmaxf = normB + DD;
    unsigned* gmax_u = (unsigned*)(gmaxf + BB);
    float* z      = (float*)(gmax_u + BB);
    float* pooled = z + BB;
    float* t1     = pooled + (size_t)BB * DD;

    const int TB = 256;
    auto blocks = [](size_t n, int tb) { return (unsigned)((n + tb - 1) / tb); };

    // zero accumulators (agg,cnt contiguous; z,pooled contiguous)
    k_fill_f32<<<blocks(ND + NPAD, TB), TB, 0, stream>>>(agg, (int)(ND + NPAD), 0.f);
    k_fill_f32<<<1, 128, 0, stream>>>(stats, 2 * DD, 0.f);
    k_fill_f32<<<blocks(BB + (size_t)BB * DD, TB), TB, 0, stream>>>(z, BB + BB * DD, 0.f);
    k_fill_u32<<<1, BB, 0, stream>>>(gmax_u, BB, 0x007FFFFFu);  // enc(-inf)

    // edge scatter mean-aggregation
    k_scatter<<<blocks((size_t)EE * DD, TB), TB, 0, stream>>>(x, ei, agg, cnt);
    k_mean<<<blocks(ND, TB), TB, 0, stream>>>(agg, cnt);

    // WMMA: hpre = mean@Wl + x@Wr + bl ; xres = x@Wres
    {
        const int waves = (NN / 16) * (DD / 16);   // 12500
        k_linear<<<blocks((size_t)waves * 32, 128), 128, 0, stream>>>(agg, x, Wl, bl, Wr,
                                                                      Wres, hpre, xres);
    }

    // batchnorm over axis 0
    k_colstats<<<256, 256, 0, stream>>>(hpre, stats);
    k_normfin<<<1, 64, 0, stream>>>(stats, gamma, beta, normA, normB);
    k_hfinal<<<blocks(ND, TB), TB, 0, stream>>>(hpre, xres, normA, normB, bres, h);

    // WMMA: gate scalar per node
    {
        const int waves = NN / 16;                 // 3125
        k_gate<<<blocks((size_t)waves * 32, 128), 128, 0, stream>>>(h, gW1, gb1, gW2, gb2, g);
    }

    // softmax attention pooling per graph
    k_gmax<<<blocks(NN, TB), TB, 0, stream>>>(g, batch, gmax_u);
    k_gmaxdec<<<1, BB, 0, stream>>>(gmax_u, gmaxf);
    k_zsum<<<blocks(NN, TB), TB, 0, stream>>>(g, batch, gmaxf, z);
    k_pool<<<blocks(ND, TB), TB, 0, stream>>>(g, batch, gmaxf, z, h, pooled);

    // head
    {
        const int waves = (BB / 16) * (HH / 16);   // 64
        k_p1<<<blocks((size_t)waves * 32, 128), 128, 0, stream>>>(pooled, Wp1, bp1, t1);
    }
    k_p2<<<1, 256, 0, stream>>>(t1, Wp2, bp2, out);
}